// UnimolEmbedding_47579647705266
// MI455X (gfx1250) — compile-verified
//
#include <hip/hip_runtime.h>

typedef __attribute__((ext_vector_type(2))) float v2f;
typedef __attribute__((ext_vector_type(8))) float v8f;

#define Bn   8
#define Ln   256
#define Dm   512
#define NH   16
#define APO  128
#define VOC  128
#define NBOND 32
#define ATOMS_EMB_ELEMS (Ln * Bn * Dm)   // 1,048,576

// -0.5 * log2(e): exp(-0.5*x) == exp2(x * NEG_HALF_LOG2E)
#define NEG_HALF_LOG2E (-0.7213475204444817f)

// ---------------------------------------------------------------------------
// Kernel A: atoms_emb[l, b, d] = atype_emb[atoms[b,l]] + chiral_emb[chirals[b,l]]
// ---------------------------------------------------------------------------
__global__ void __launch_bounds__(256)
atoms_emb_kernel(const int* __restrict__ atoms,
                 const int* __restrict__ chirals,
                 const float* __restrict__ atype_emb,
                 const float* __restrict__ chiral_emb,
                 float* __restrict__ out) {
    int v   = blockIdx.x * blockDim.x + threadIdx.x;  // float4 index
    int k4  = v & (Dm / 4 - 1);                       // 0..127
    int lb  = v >> 7;                                 // (l_idx*8 + b_idx)
    int li  = lb >> 3;
    int bi  = lb & 7;
    int a   = atoms[bi * Ln + li];
    int c   = chirals[bi * Ln + li];
    const float4* ar = (const float4*)(atype_emb  + (size_t)a * Dm);
    const float4* cr = (const float4*)(chiral_emb + (size_t)c * Dm);
    float4 x = ar[k4];
    float4 y = cr[k4];
    float4 r;
    r.x = x.x + y.x; r.y = x.y + y.y; r.z = x.z + y.z; r.w = x.w + y.w;
    ((float4*)out)[v] = r;
}

// ---------------------------------------------------------------------------
// Kernel B: apairs[b, h, i, j]
//   One wave32 per 16(j) x 16(head) tile. K=128 in 32 steps of
//   V_WMMA_F32_16X16X4_F32 (D = A(16x4 gauss) * B(4x16 lin_w) + C).
//
// A-matrix f32 16x4 layout (ISA 7.12.2): lane L (L<16): M=L, v0=K0, v1=K1;
//                                        lane L+16:     M=L, v0=K2, v1=K3.
// B-matrix 4x16 mirrors it: lane L: N=L&15, v0=lin_w[k0][n], v1=lin_w[k0+1][n].
// C/D: VGPR r, lanes 0-15: M=r, N=lane; lanes 16-31: M=r+8, N=lane-16.
//
// All block-invariant small tables (lin_w 8KB, bond_emb 2KB, lin_b,
// std-derived constants) live in LDS; the hot loop's only VMEM traffic is
// the two mandatory float2 pair-table gathers (L2-resident, 8MB each).
// ---------------------------------------------------------------------------
__global__ void __launch_bounds__(256)
apairs_kernel(const int*   __restrict__ atoms,
              const float* __restrict__ coords,
              const int*   __restrict__ bonds,
              const float* __restrict__ w_tab,
              const float* __restrict__ b_tab,
              const float* __restrict__ means,
              const float* __restrict__ stds,
              const float* __restrict__ bond_emb,
              const float* __restrict__ lin_w,
              const float* __restrict__ lin_b,
              float* __restrict__ out_apairs) {
    __shared__ float s_linw[APO * NH];    // 8 KB: lin_w, k-major [k][h]
    __shared__ float s_bond[NBOND * NH];  // 2 KB: bond_emb [bond][h]
    __shared__ float s_linb[NH];          // 64 B
    __shared__ float s_inv[APO];          // 1/(|std|+1e-5)
    __shared__ float s_mA[APO];           // mean * inv
    __shared__ float s_nA[APO];           // inv / sqrt(2*pi)

    const int tid = threadIdx.x;

    // ---- block prologue: stage block-invariant tables in LDS ------------
    {
        float4* dv = (float4*)s_linw;
        const float4* sv = (const float4*)lin_w;   // 512 float4s
        dv[tid]       = sv[tid];
        dv[tid + 256] = sv[tid + 256];
    }
    if (tid < APO) {
        ((float4*)s_bond)[tid] = ((const float4*)bond_emb)[tid];  // 128 f4
        const float sd = __builtin_fabsf(stds[tid]) + 1e-5f;
        const float iv = __builtin_amdgcn_rcpf(sd);               // v_rcp_f32
        s_inv[tid] = iv;
        s_mA[tid]  = means[tid] * iv;
        s_nA[tid]  = iv * 0.3989422804014327f;                    // 1/sqrt(2pi)
    }
    if (tid < NH) s_linb[tid] = lin_b[tid];
    __syncthreads();                                  // EXEC all-1 after

    // ---- tile coordinates ----------------------------------------------
    const int lane = tid & 31;
    const int wave = tid >> 5;
    const int tile = blockIdx.x * 8 + wave;           // 32768 tiles total

    const int jt = tile & 15;                         // j-tile (l/16 = 16)
    const int i  = (tile >> 4) & (Ln - 1);
    const int b  = tile >> 12;                        // /(16*256)
    const int j0 = jt * 16;

    const int m    = lane & 15;                       // A row / head index
    const int hi2  = lane >> 4;                       // half: 0 or 1
    const int koff = hi2 * 2;                         // K offset inside chunk
    const int j    = j0 + m;

    // pair index + distance for row m (lanes L and L+16 duplicate row m: OK)
    const int a_i  = atoms[b * Ln + i];
    const int a_j  = atoms[b * Ln + j];
    const int pidx = a_j * VOC + a_i;
    const float* wrow = w_tab + (size_t)pidx * APO;
    const float* brow = b_tab + (size_t)pidx * APO;

    const float* ci = coords + (size_t)(b * Ln + i) * 3;
    const float* cj = coords + (size_t)(b * Ln + j) * 3;
    const float dx = cj[0] - ci[0];
    const float dy = cj[1] - ci[1];
    const float dz = cj[2] - ci[2];
    const float s  = dx * dx + dy * dy + dz * dz;
    const float dist = (s > 0.0f) ? __builtin_amdgcn_sqrtf(s) : 0.0f;

    v8f acc = {};                                     // f32 16x16 accumulator

    #pragma unroll 4
    for (int ch = 0; ch < APO / 4; ++ch) {
        const int k0 = ch * 4 + koff;
        // the only VMEM in the hot loop: contiguous float2 table gathers
        const float2 w2 = *(const float2*)(wrow + k0);
        const float2 t2 = *(const float2*)(brow + k0);
        // block-invariant constants from LDS (DS pipe)
        const float2 iv2 = *(const float2*)(s_inv + k0);
        const float2 ma2 = *(const float2*)(s_mA + k0);
        const float2 na2 = *(const float2*)(s_nA + k0);

        const float t0 = __builtin_fmaf(w2.x, dist, t2.x);
        const float t1 = __builtin_fmaf(w2.y, dist, t2.y);
        const float u0 = __builtin_fmaf(t0, iv2.x, -ma2.x);
        const float u1 = __builtin_fmaf(t1, iv2.y, -ma2.y);
        // exp(-0.5*u^2) = exp2(u^2 * -0.5*log2(e)): 2 muls + v_exp_f32
        const float g0 = __builtin_amdgcn_exp2f(u0 * u0 * NEG_HALF_LOG2E) * na2.x;
        const float g1 = __builtin_amdgcn_exp2f(u1 * u1 * NEG_HALF_LOG2E) * na2.y;

        v2f avec; avec.x = g0; avec.y = g1;
        v2f bvec;
        bvec.x = s_linw[(k0 + 0) * NH + m];           // B: K=k0,   N=m
        bvec.y = s_linw[(k0 + 1) * NH + m];           // B: K=k0+1, N=m

        // D = A * B + C, native f32 WMMA
        acc = __builtin_amdgcn_wmma_f32_16x16x4_f32(
                  /*neg_a=*/false, avec,
                  /*neg_b=*/false, bvec,
                  /*c_mod=*/(short)0, acc,
                  /*reuse_a=*/false, /*reuse_b=*/false);
    }

    // ---- epilogue: + lin_b[h] + bond_emb[bonds,h]; pad-mask -> -inf ----
    const float lbv = s_linb[m];                      // head h == m for C lanes
    const int*  bondrow = bonds + ((size_t)(b * Ln + i)) * Ln;
    const int*  atomrow = atoms + b * Ln;
    float* dst = out_apairs + (((size_t)(b * NH + m) * Ln + i) * Ln);

    const int jbase = j0 + hi2 * 8;                   // rows r -> j = jbase + r
    const int4 bd0 = *(const int4*)(bondrow + jbase);
    const int4 bd1 = *(const int4*)(bondrow + jbase + 4);
    const int4 at0 = *(const int4*)(atomrow + jbase);
    const int4 at1 = *(const int4*)(atomrow + jbase + 4);
    const int bd[8] = {bd0.x, bd0.y, bd0.z, bd0.w, bd1.x, bd1.y, bd1.z, bd1.w};
    const int at[8] = {at0.x, at0.y, at0.z, at0.w, at1.x, at1.y, at1.z, at1.w};

    float vals[8];
    #pragma unroll
    for (int r = 0; r < 8; ++r) {
        float v = acc[r] + lbv + s_bond[bd[r] * NH + m];   // DS gather
        if (at[r] == 0) v = -__builtin_inff();             // PAD mask (key dim)
        vals[r] = v;
    }
    float4 o0, o1;
    o0.x = vals[0]; o0.y = vals[1]; o0.z = vals[2]; o0.w = vals[3];
    o1.x = vals[4]; o1.y = vals[5]; o1.z = vals[6]; o1.w = vals[7];
    *(float4*)(dst + jbase)     = o0;
    *(float4*)(dst + jbase + 4) = o1;
}

// ---------------------------------------------------------------------------
extern "C" void kernel_launch(void* const* d_in, const int* in_sizes, int n_in,
                              void* d_out, int out_size, void* d_ws, size_t ws_size,
                              hipStream_t stream) {
    const int*   atoms      = (const int*)  d_in[0];
    const int*   chirals    = (const int*)  d_in[1];
    const float* coords     = (const float*)d_in[2];
    const int*   bonds      = (const int*)  d_in[3];
    const float* atype_emb  = (const float*)d_in[4];
    const float* chiral_emb = (const float*)d_in[5];
    const float* w_tab      = (const float*)d_in[6];
    const float* b_tab      = (const float*)d_in[7];
    const float* means      = (const float*)d_in[8];
    const float* stds       = (const float*)d_in[9];
    const float* bond_emb   = (const float*)d_in[10];
    const float* lin_w      = (const float*)d_in[11];
    const float* lin_b      = (const float*)d_in[12];

    float* out_atoms  = (float*)d_out;                   // [l, b, d]  = 1,048,576
    float* out_apairs = out_atoms + ATOMS_EMB_ELEMS;     // [b, nh, l, l] = 8,388,608

    // Kernel A: 262,144 float4 elements / 256 threads
    atoms_emb_kernel<<<ATOMS_EMB_ELEMS / 4 / 256, 256, 0, stream>>>(
        atoms, chirals, atype_emb, chiral_emb, out_atoms);

    // Kernel B: 8 * 256 * (256/16) = 32768 tiles; 8 waves (256 thr) per block
    apairs_kernel<<<32768 / 8, 256, 0, stream>>>(
        atoms, coords, bonds, w_tab, b_tab, means, stds,
        bond_emb, lin_w, lin_b, out_apairs);
}